// Regress_graph_67577015435850
// MI455X (gfx1250) — compile-verified
//
#include <hip/hip_runtime.h>
#include <hip/hip_bf16.h>
#include <math.h>

typedef __attribute__((ext_vector_type(2))) float v2f;
typedef __attribute__((ext_vector_type(8))) float v8f;

#define HID 128

// ---------------- degree / normalization ----------------

__global__ void init_deg(float* deg, int n) {
    int i = blockIdx.x * blockDim.x + threadIdx.x;
    if (i < n) deg[i] = 1.0f;          // self loop contributes 1
}

__global__ void accum_deg(const int* __restrict__ dst, float* deg, int E) {
    int e = blockIdx.x * blockDim.x + threadIdx.x;
    if (e < E) atomicAdd(&deg[dst[e]], 1.0f);
}

__global__ void finish_dinv(float* deg, int n) {
    int i = blockIdx.x * blockDim.x + threadIdx.x;
    if (i < n) deg[i] = rsqrtf(deg[i]);   // in-place: deg -> dinv
}

__global__ void edge_norm(const int* __restrict__ src, const int* __restrict__ dst,
                          const float* __restrict__ dinv, float* __restrict__ enorm, int E) {
    int e = blockIdx.x * blockDim.x + threadIdx.x;
    if (e < E) enorm[e] = dinv[src[e]] * dinv[dst[e]];
}

// ---------------- fp32 WMMA GEMM: C[N,128] = A[N,128] @ W[128,128] ----------------
// One wave -> one 16x16 output tile, K stepped by 4 (V_WMMA_F32_16X16X4_F32).
// A frag (16x4): lane<16 holds {M=l16,K=k0..k0+1}, lane>=16 holds {M=l16,K=k0+2..k0+3}.
// B frag (4x16): lane<16 holds {K=k0..k0+1,N=l16}, lane>=16 holds {K=k0+2..k0+3,N=l16}.
// C/D: VGPR r -> M = r + 8*(lane>=16), N = l16.

__global__ __launch_bounds__(128) void gemm_wmma_f32(const float* __restrict__ A,
                                                     const float* __restrict__ W,
                                                     float* __restrict__ C) {
    const int wave = threadIdx.x >> 5;
    const int lane = threadIdx.x & 31;
    const int half = lane >> 4;
    const int l16  = lane & 15;
    const int row0 = blockIdx.x * 16;
    const int col0 = (blockIdx.y * 4 + wave) * 16;

    const float* __restrict__ arow = A + (size_t)(row0 + l16) * HID;
    v8f acc = {};
#pragma unroll
    for (int k0 = 0; k0 < HID; k0 += 4) {
        const int k = k0 + 2 * half;
        v2f a = { arow[k], arow[k + 1] };
        v2f b = { W[(size_t)k * HID + col0 + l16],
                  W[(size_t)(k + 1) * HID + col0 + l16] };
        acc = __builtin_amdgcn_wmma_f32_16x16x4_f32(
            /*neg_a=*/false, a, /*neg_b=*/false, b,
            /*c_mod=*/(short)0, acc, /*reuse_a=*/false, /*reuse_b=*/false);
    }
#pragma unroll
    for (int r = 0; r < 8; ++r) {
        C[(size_t)(row0 + r + 8 * half) * HID + col0 + l16] = acc[r];
    }
}

// ---------------- edge scatter (L2-resident fp32 atomics) ----------------
// One wave per edge: 32 lanes x float4 = 128 features.

__global__ __launch_bounds__(256) void scatter_edges(const float* __restrict__ hw,
                                                     const float* __restrict__ enorm,
                                                     const int* __restrict__ src,
                                                     const int* __restrict__ dst,
                                                     float* __restrict__ agg, int E) {
    int gt = blockIdx.x * 256 + threadIdx.x;
    int e = gt >> 5;
    int lane = gt & 31;
    if (e >= E) return;
    int s = src[e], d = dst[e];
    float nrm = enorm[e];
    const float4* hv = reinterpret_cast<const float4*>(hw + (size_t)s * HID);
    float4 v = hv[lane];
    float* ap = agg + (size_t)d * HID + lane * 4;
    atomicAdd(ap + 0, nrm * v.x);
    atomicAdd(ap + 1, nrm * v.y);
    atomicAdd(ap + 2, nrm * v.z);
    atomicAdd(ap + 3, nrm * v.w);
}

__global__ void zero_f32(float* p, long n) {
    long i = (long)blockIdx.x * blockDim.x + threadIdx.x;
    if (i < n) p[i] = 0.0f;
}

// ---------------- ELU + bias + fused self-loop term ----------------
// h_out = elu(agg + dinv[n]^2 * hw + b)

__global__ void elu_bias(const float* __restrict__ agg, const float* __restrict__ hw,
                         const float* __restrict__ dinv, const float* __restrict__ bias,
                         float* __restrict__ hout, int nNodes) {
    long i = (long)blockIdx.x * blockDim.x + threadIdx.x;
    if (i >= (long)nNodes * HID) return;
    int n = (int)(i >> 7);
    int f = (int)(i & (HID - 1));
    float di = dinv[n];
    float x = agg[i] + di * di * hw[i] + bias[f];
    hout[i] = x > 0.0f ? x : expm1f(x);
}

// ---------------- masked segment-max pool + final dot ----------------
// batch is sorted: binary-search per-graph bounds; 128 threads = 128 features.

__global__ __launch_bounds__(128) void pool_dot(const float* __restrict__ h,
                                                const int* __restrict__ mask,
                                                const int* __restrict__ batch,
                                                const float* __restrict__ w,
                                                const float* __restrict__ b,
                                                float* __restrict__ out, int M) {
    int g = blockIdx.x;
    int f = threadIdx.x;

    int lo = 0, hi = M;
    while (lo < hi) { int mid = (lo + hi) >> 1; if (batch[mid] < g) lo = mid + 1; else hi = mid; }
    int beg = lo;
    hi = M;
    while (lo < hi) { int mid = (lo + hi) >> 1; if (batch[mid] < g + 1) lo = mid + 1; else hi = mid; }
    int end = lo;

    float m = -__builtin_huge_valf();        // segment_max identity for empty segments
    for (int i = beg; i < end; ++i) {
        m = fmaxf(m, h[(size_t)mask[i] * HID + f]);
    }
    float v = m * w[f];

    __shared__ float red[HID];
    red[f] = v;
    __syncthreads();
#pragma unroll
    for (int s = HID / 2; s > 0; s >>= 1) {
        if (f < s) red[f] += red[f + s];
        __syncthreads();
    }
    if (f == 0) out[g] = red[0] + b[0];
}

// ---------------- host side ----------------

extern "C" void kernel_launch(void* const* d_in, const int* in_sizes, int n_in,
                              void* d_out, int out_size, void* d_ws, size_t ws_size,
                              hipStream_t stream) {
    const float* x      = (const float*)d_in[0];
    const int*   ei     = (const int*)d_in[1];
    const int*   mask   = (const int*)d_in[2];
    const int*   batch  = (const int*)d_in[3];
    const float* conv_w = (const float*)d_in[4];
    const float* conv_b = (const float*)d_in[5];
    const float* lt1_w  = (const float*)d_in[6];
    const float* lt1_b  = (const float*)d_in[7];
    float* out = (float*)d_out;

    const int nNodes = in_sizes[0] / HID;     // 50000
    const int E      = in_sizes[1] / 2;       // 800000
    const int M      = in_sizes[2];           // 25000
    const int G      = out_size;              // 64
    const size_t NH  = (size_t)nNodes * HID;

    const int* src = ei;
    const int* dst = ei + E;

    // workspace carve-up (fp32): dinv | enorm | h | hw | agg
    float* wsf   = (float*)d_ws;
    float* dinv  = wsf;
    float* enorm = dinv + ((nNodes + 255) / 256) * 256;
    float* hbuf  = enorm + ((E + 255) / 256) * 256;
    float* hwb   = hbuf + NH;
    float* aggb  = hwb + NH;

    const int T = 256;
    const int nodeBlocks = (nNodes + T - 1) / T;
    const int edgeBlocks = (E + T - 1) / T;
    const int elemBlocks = (int)((NH + T - 1) / T);
    const int scatBlocks = (int)(((size_t)E * 32 + T - 1) / T);

    // 1) degrees -> dinv -> per-edge norms (reused across all 3 layers)
    init_deg<<<nodeBlocks, T, 0, stream>>>(dinv, nNodes);
    accum_deg<<<edgeBlocks, T, 0, stream>>>(dst, dinv, E);
    finish_dinv<<<nodeBlocks, T, 0, stream>>>(dinv, nNodes);
    edge_norm<<<edgeBlocks, T, 0, stream>>>(src, dst, dinv, enorm, E);

    // 2) three GCN layers
    dim3 gemmGrid((nNodes + 15) / 16, 2);     // 2 * 4 waves = 8 column tiles of 16
    for (int l = 0; l < 3; ++l) {
        const float* hin = (l == 0) ? x : hbuf;
        const float* Wl  = conv_w + (size_t)l * HID * HID;
        const float* bl  = conv_b + (size_t)l * HID;

        gemm_wmma_f32<<<gemmGrid, 128, 0, stream>>>(hin, Wl, hwb);
        zero_f32<<<elemBlocks, T, 0, stream>>>(aggb, (long)NH);
        scatter_edges<<<scatBlocks, T, 0, stream>>>(hwb, enorm, src, dst, aggb, E);
        elu_bias<<<elemBlocks, T, 0, stream>>>(aggb, hwb, dinv, bl, hbuf, nNodes);
    }

    // 3) masked max-pool per graph + dot with lt1_w
    pool_dot<<<G, HID, 0, stream>>>(hbuf, mask, batch, lt1_w, lt1_b, out, M);
}